// GCNLayer_90331752169530
// MI455X (gfx1250) — compile-verified
//
#include <hip/hip_runtime.h>

typedef __attribute__((ext_vector_type(2))) float v2f;
typedef __attribute__((ext_vector_type(8))) float v8f;

#define NN   512
#define FD   128
#define GP2  144                   // float2 stride per k-pair row (padded: 128+16)
#define XW_FLOATS (256 * 2 * GP2)  // 256 k-pairs * 288 floats = 73728 floats = 288 KB

// One workgroup per batch. 1024 threads = 32 waves; wave w owns rows [16w, 16w+16).
__global__ __launch_bounds__(1024)
void gcn_fused_kernel(const float* __restrict__ adj,
                      const float* __restrict__ x,
                      const float* __restrict__ W,
                      const float* __restrict__ bias,
                      float* __restrict__ out)
{
    extern __shared__ float smem[];
    float* xw      = smem;               // [256 k-pairs][GP2 float2] interleaved (k, k+1)
    float* dnorm_s = smem + XW_FLOATS;   // [512]
    float* bias_s  = dnorm_s + NN;       // [128]

    const int b    = blockIdx.x;
    const int tid  = threadIdx.x;
    const int wave = tid >> 5;
    const int lane = tid & 31;
    const int ln   = lane & 15;   // column / row-in-half position
    const int hh   = lane >> 4;   // which half of the wave (K split)
    const int h2   = hh * 2;

    const float* adjb = adj + (size_t)b * NN * NN;
    const float* xb   = x   + (size_t)b * NN * FD;
    float*       outb = out + (size_t)b * NN * FD;

    const int mtile = wave * 16;

    // ---------------- Phase 0: degree + dnorm (per-wave rows), bias ----------------
    if (tid < FD) bias_s[tid] = bias[tid];

    for (int r = 0; r < 16; ++r) {
        const int row = mtile + r;
        const float* rp = adjb + (size_t)row * NN;
        float s = 0.f;
        #pragma unroll
        for (int i = 0; i < 16; ++i) s += rp[lane + i * 32];   // coalesced row sum
        #pragma unroll
        for (int off = 16; off > 0; off >>= 1) s += __shfl_xor(s, off, 32);
        if (lane == 0) dnorm_s[row] = (s > 0.f) ? rsqrtf(s) : 0.f;
    }
    __syncthreads();

    // dnorm for this wave's 8 C-rows (VGPR j -> row mtile + hh*8 + j); reused in
    // both epilogues — load once into registers to avoid serialized LDS round-trips.
    float dn[8];
    #pragma unroll
    for (int j = 0; j < 8; ++j) dn[j] = dnorm_s[mtile + hh * 8 + j];

    // ---------------- Phase 1: xw = dnorm ⊙ (x @ W) -> LDS ----------------
    // A frag (16x4 f32): lane ln row mtile+ln, K = k+h2, k+h2+1  (contiguous float2).
    // One A load feeds all 8 column tiles (8 live accumulators).
    {
        v8f acc[8];
        #pragma unroll
        for (int ct = 0; ct < 8; ++ct)
            acc[ct] = (v8f){0.f,0.f,0.f,0.f,0.f,0.f,0.f,0.f};

        const float* xrow = xb + (size_t)(mtile + ln) * FD;
        for (int k = 0; k < FD; k += 4) {
            const int kk = k + h2;
            v2f a = *(const v2f*)(xrow + kk);
            const float* wbase = W + kk * FD + ln;
            #pragma unroll
            for (int ct = 0; ct < 8; ++ct) {
                v2f bf;
                bf.x = wbase[ct * 16];
                bf.y = wbase[ct * 16 + FD];
                acc[ct] = __builtin_amdgcn_wmma_f32_16x16x4_f32(
                              false, a, false, bf, (short)0, acc[ct], false, false);
            }
        }
        // Store into k-pair-interleaved LDS: float offset = (m>>1)*2*GP2 + 2*g + (m&1)
        #pragma unroll
        for (int ct = 0; ct < 8; ++ct) {
            const int g = ct * 16 + ln;
            #pragma unroll
            for (int j = 0; j < 8; ++j) {
                const int m = mtile + hh * 8 + j;
                xw[(m >> 1) * (2 * GP2) + 2 * g + (m & 1)] = acc[ct][j] * dn[j];
            }
        }
    }
    __syncthreads();

    // ---------------- Phase 2: y = relu(dnorm ⊙ (adj @ xw) + bias) ----------------
    {
        v8f acc[8];
        #pragma unroll
        for (int ct = 0; ct < 8; ++ct)
            acc[ct] = (v8f){0.f,0.f,0.f,0.f,0.f,0.f,0.f,0.f};

        const float* arow = adjb + (size_t)(mtile + ln) * NN;   // A rows (this wave's tile)
        for (int k = 0; k < NN; k += 4) {
            const int kk = k + h2;
            v2f a = *(const v2f*)(arow + kk);                   // 8B coalesced per lane
            // B frags: pair (K=kk, kk+1) is one aligned 8B LDS load, conflict-free
            const float* bbase = xw + ((k >> 1) + hh) * (2 * GP2) + 2 * ln;
            #pragma unroll
            for (int ct = 0; ct < 8; ++ct) {
                v2f bf = *(const v2f*)(bbase + 32 * ct);
                acc[ct] = __builtin_amdgcn_wmma_f32_16x16x4_f32(
                              false, a, false, bf, (short)0, acc[ct], false, false);
            }
        }

        // Epilogue: all scale factors already in registers (dn[], bias preloaded).
        #pragma unroll
        for (int ct = 0; ct < 8; ++ct) {
            const int col = ct * 16 + ln;
            const float bcol = bias_s[col];
            #pragma unroll
            for (int j = 0; j < 8; ++j) {
                const int nrow = mtile + hh * 8 + j;
                float v = acc[ct][j] * dn[j] + bcol;
                outb[(size_t)nrow * FD + col] = fmaxf(v, 0.f);
            }
        }
    }
}

extern "C" void kernel_launch(void* const* d_in, const int* in_sizes, int n_in,
                              void* d_out, int out_size, void* d_ws, size_t ws_size,
                              hipStream_t stream) {
    const float* adj  = (const float*)d_in[0];
    const float* x    = (const float*)d_in[1];
    const float* W    = (const float*)d_in[2];
    const float* bias = (const float*)d_in[3];

    const size_t smem_bytes = (size_t)(XW_FLOATS + NN + FD) * sizeof(float); // ~297.5 KB

    static bool attr_set = false;  // idempotent, not stream-dependent; safe under capture
    if (!attr_set) {
        hipFuncSetAttribute((const void*)gcn_fused_kernel,
                            hipFuncAttributeMaxDynamicSharedMemorySize,
                            (int)smem_bytes);
        attr_set = true;
    }

    gcn_fused_kernel<<<128, 1024, smem_bytes, stream>>>(adj, x, W, bias, (float*)d_out);
}